// TripletLossWithGL_6536940224734
// MI455X (gfx1250) — compile-verified
//
#include <hip/hip_runtime.h>
#include <hip/hip_bf16.h>
#include <math.h>

typedef __attribute__((ext_vector_type(2))) float v2f;
typedef __attribute__((ext_vector_type(8))) float v8f;

#define NROWS 4096
#define KDIM  256
#define BIGF  10000.0f
#define MARGINF 0.3f
#define INF_BITS 0x7F800000u

// ---------------- init workspace ----------------
__global__ __launch_bounds__(256) void tl_init_kernel(unsigned* __restrict__ rowmax,
                                                      unsigned* __restrict__ minG,
                                                      unsigned* __restrict__ minL,
                                                      float* __restrict__ lossAcc,
                                                      int* __restrict__ cntAcc,
                                                      int* __restrict__ hist) {
    int i = blockIdx.x * 256 + threadIdx.x;
    if (i < NROWS) {
        rowmax[i] = 0u;          // dist >= 0, diagonal is always in eq-set
        minG[i]   = INF_BITS;
        minL[i]   = INF_BITS;
    }
    if (i == 0) {
        *lossAcc = 0.0f;
        *cntAcc  = 0;
        hist[0] = hist[1] = hist[2] = hist[3] = 0;
    }
}

// ---------------- row squared norms + label histogram ----------------
__global__ __launch_bounds__(256) void tl_sqnorm_kernel(const float* __restrict__ X,
                                                        const int* __restrict__ T,
                                                        float* __restrict__ sq,
                                                        int* __restrict__ hist) {
    const int wave = threadIdx.x >> 5;
    const int lane = threadIdx.x & 31;
    const int row  = blockIdx.x * 8 + wave;   // 512 blocks * 8 waves = 4096 rows
    const float* xr = X + (size_t)row * KDIM;
    float s = 0.0f;
#pragma unroll
    for (int c = lane; c < KDIM; c += 32) {
        float v = xr[c];
        s = fmaf(v, v, s);
    }
#pragma unroll
    for (int off = 16; off >= 1; off >>= 1) s += __shfl_xor(s, off, 32);
    if (lane == 0) {
        sq[row] = s;
        atomicAdd(&hist[T[row] & 3], 1);
    }
}

// ---------------- fused Gram-WMMA + dist + masked row reductions ----------------
// Block = 256 threads = 8 waves. Block tile = 128 rows x 128 cols.
// Wave w covers rows [i0, i0+16) x cols [j0, j0+128) as 8 accumulators of
// v_wmma_f32_16x16x4_f32 over K=256.
__global__ __launch_bounds__(256) void tl_dist_wmma_kernel(const float* __restrict__ X,
                                                           const int* __restrict__ T,
                                                           const float* __restrict__ sq,
                                                           float* __restrict__ dist,
                                                           unsigned* __restrict__ rowmax,
                                                           unsigned* __restrict__ minG,
                                                           unsigned* __restrict__ minL) {
    const int wave = threadIdx.x >> 5;
    const int lane = threadIdx.x & 31;
    const int half = lane >> 4;     // 0: lanes 0-15, 1: lanes 16-31
    const int l15  = lane & 15;

    const int i0 = blockIdx.y * 128 + wave * 16;
    const int j0 = blockIdx.x * 128;

    // A-matrix 16x4 f32 layout: lane<16 holds (M=l15, K={k,k+1}); lane>=16 holds (M=l15, K={k+2,k+3})
    const float* aPtr  = X + (size_t)(i0 + l15) * KDIM + 2 * half;
    // B-matrix 4x16: symmetric layout with N in place of M; B[k][n] = X[j0+n][k]
    const float* bBase = X + (size_t)(j0 + l15) * KDIM + 2 * half;

    v8f acc[8] = {};

    for (int k = 0; k < KDIM; k += 4) {
        v2f a = *(const v2f*)(aPtr + k);
#pragma unroll
        for (int jt = 0; jt < 8; ++jt) {
            v2f b = *(const v2f*)(bBase + (size_t)jt * 16 * KDIM + k);
            // (neg_a, A, neg_b, B, c_mod, C, reuse_a, reuse_b)
            acc[jt] = __builtin_amdgcn_wmma_f32_16x16x4_f32(
                false, a, false, b, (short)0, acc[jt], false, false);
        }
    }

    // Preload column labels / norms for this lane's 8 columns
    int   tj[8];
    float sqn[8];
#pragma unroll
    for (int jt = 0; jt < 8; ++jt) {
        const int n = j0 + jt * 16 + l15;
        tj[jt]  = T[n];
        sqn[jt] = sq[n];
    }

    // C/D layout: VGPR r, lanes 0-15 -> (M=r, N=lane); lanes 16-31 -> (M=r+8, N=lane-16)
#pragma unroll
    for (int r = 0; r < 8; ++r) {
        const int   m   = i0 + half * 8 + r;   // uniform across the 16-lane half
        const int   ti  = T[m];
        const float sqm = sq[m];

        float mEq = 0.0f;                       // identity: dist>=0, diag in eq-set
        float mG  = __builtin_huge_valf();
        float mL  = __builtin_huge_valf();
#pragma unroll
        for (int jt = 0; jt < 8; ++jt) {
            const int n = j0 + jt * 16 + l15;
            float d2 = sqm + sqn[jt] - 2.0f * acc[jt][r];
            float d  = sqrtf(fmaxf(d2, 0.0f));
            dist[(size_t)m * NROWS + n] = d;
            if (tj[jt] == ti)      mEq = fmaxf(mEq, d);
            else if (tj[jt] > ti)  mG  = fminf(mG, d);
            else                   mL  = fminf(mL, d);
        }
        // reduce across the 16 lanes of this half (xor masks < 16 stay in-half)
#pragma unroll
        for (int off = 1; off < 16; off <<= 1) {
            mEq = fmaxf(mEq, __shfl_xor(mEq, off, 32));
            mG  = fminf(mG,  __shfl_xor(mG,  off, 32));
            mL  = fminf(mL,  __shfl_xor(mL,  off, 32));
        }
        if (l15 == 0) {
            // dist >= 0 -> f32 bit pattern is order-preserving as unsigned
            atomicMax(&rowmax[m], __float_as_uint(mEq));
            atomicMin(&minG[m],   __float_as_uint(mG));
            atomicMin(&minL[m],   __float_as_uint(mL));
        }
    }
}

// ---------------- per-row loss terms + cnt, reduced into scalars ----------------
__global__ __launch_bounds__(256) void tl_finalize_kernel(const unsigned* __restrict__ rowmax,
                                                          const unsigned* __restrict__ minG,
                                                          const unsigned* __restrict__ minL,
                                                          const int* __restrict__ T,
                                                          const int* __restrict__ hist,
                                                          float* __restrict__ lossAcc,
                                                          int* __restrict__ cntAcc) {
    const int i = blockIdx.x * 256 + threadIdx.x;
    float hard = __uint_as_float(rowmax[i]);
    unsigned g = minG[i], l = minL[i];
    float ag = (g >= INF_BITS) ? BIGF : __uint_as_float(g);
    float al = (l >= INF_BITS) ? BIGF : __uint_as_float(l);
    float term = fmaxf(hard - fabsf(ag - al) + MARGINF, 0.0f);
    int c = (hist[T[i] & 3] == 1) ? 1 : 0;
#pragma unroll
    for (int off = 16; off >= 1; off >>= 1) {
        term += __shfl_xor(term, off, 32);
        c    += __shfl_xor(c, off, 32);
    }
    if ((threadIdx.x & 31) == 0) {
        atomicAdd(lossAcc, term);
        atomicAdd(cntAcc, c);
    }
}

__global__ void tl_writeout_kernel(const float* __restrict__ lossAcc,
                                   const int* __restrict__ cntAcc,
                                   float* __restrict__ out) {
    out[0] = *lossAcc * (1.0f / (float)NROWS);
    out[(size_t)1 + (size_t)NROWS * NROWS] = (float)(*cntAcc);
}

extern "C" void kernel_launch(void* const* d_in, const int* in_sizes, int n_in,
                              void* d_out, int out_size, void* d_ws, size_t ws_size,
                              hipStream_t stream) {
    const float* X = (const float*)d_in[0];   // (4096, 256) f32
    const int*   T = (const int*)d_in[1];     // (4096,)   i32
    float* out = (float*)d_out;               // [loss | dist(4096^2) | cnt]

    char* ws = (char*)d_ws;
    float*    sq      = (float*)(ws + 0);          // 4096 f32
    unsigned* rowmax  = (unsigned*)(ws + 16384);   // 4096 u32
    unsigned* minG    = (unsigned*)(ws + 32768);   // 4096 u32
    unsigned* minL    = (unsigned*)(ws + 49152);   // 4096 u32
    int*      hist    = (int*)(ws + 65536);        // 4 i32
    float*    lossAcc = (float*)(ws + 65536 + 16); // 1 f32
    int*      cntAcc  = (int*)(ws + 65536 + 20);   // 1 i32

    tl_init_kernel<<<16, 256, 0, stream>>>(rowmax, minG, minL, lossAcc, cntAcc, hist);
    tl_sqnorm_kernel<<<512, 256, 0, stream>>>(X, T, sq, hist);

    dim3 grid(NROWS / 128, NROWS / 128);  // 32 x 32 blocks, 128x128 tile each
    tl_dist_wmma_kernel<<<grid, 256, 0, stream>>>(X, T, sq, out + 1, rowmax, minG, minL);

    tl_finalize_kernel<<<16, 256, 0, stream>>>(rowmax, minG, minL, T, hist, lossAcc, cntAcc);
    tl_writeout_kernel<<<1, 1, 0, stream>>>(lossAcc, cntAcc, out);
}